// SpikingNeuralNet_67130338836636
// MI455X (gfx1250) — compile-verified
//
#include <hip/hip_runtime.h>
#include <hip/hip_bf16.h>
#include <stdint.h>
#include <stddef.h>

#define NN      2048
#define TSTEPS  4000
#define MAXSP   512
#define ALPHA_C 0.01f
#define VRESET  1.0f

// CDNA5 async global->LDS path (ASYNCcnt). Guarded: host pass and older
// toolchains fall back to direct global reads of the w row.
#if defined(__AMDGCN__) && __has_builtin(__builtin_amdgcn_global_load_async_to_lds_b128) && __has_builtin(__builtin_amdgcn_s_wait_asynccnt)
#define USE_ASYNC 1
typedef int v4i_t __attribute__((ext_vector_type(4)));
typedef __attribute__((address_space(1))) v4i_t* g_v4i_ptr;   // global (HIP "__device__")
typedef __attribute__((address_space(3))) v4i_t* l_v4i_ptr;   // LDS    (HIP "__shared__")
#else
#define USE_ASYNC 0
#endif

__device__ __forceinline__ uint32_t rotl32(uint32_t v, int r) {
  return (v << r) | (v >> (32 - r));
}

// Bit-exact JAX threefry2x32 (20 rounds, rotation schedule {13,15,26,6,17,29,16,24})
__device__ __forceinline__ void threefry2x32(uint32_t k0, uint32_t k1,
                                             uint32_t x0, uint32_t x1,
                                             uint32_t& o0, uint32_t& o1) {
  const uint32_t k2 = k0 ^ k1 ^ 0x1BD11BDAu;
  uint32_t a = x0 + k0, b = x1 + k1;
#define TFR(r) a += b; b = rotl32(b, (r)); b ^= a;
  TFR(13) TFR(15) TFR(26) TFR(6)  a += k1; b += k2 + 1u;
  TFR(17) TFR(29) TFR(16) TFR(24) a += k2; b += k0 + 2u;
  TFR(13) TFR(15) TFR(26) TFR(6)  a += k0; b += k1 + 3u;
  TFR(17) TFR(29) TFR(16) TFR(24) a += k1; b += k2 + 4u;
  TFR(13) TFR(15) TFR(26) TFR(6)  a += k2; b += k0 + 5u;
#undef TFR
  o0 = a; o1 = b;
}

// JAX uniform [0,1): (bits>>9 | 0x3f800000) bitcast - 1.0
__device__ __forceinline__ float u01(uint32_t bits) {
  return __uint_as_float((bits >> 9) | 0x3f800000u) - 1.0f;
}

// jax.nn.softplus(x) = max(x,0) + log1p(exp(-|x|))
__device__ __forceinline__ float softplus(float x) {
  return fmaxf(x, 0.0f) + log1pf(expf(-fabsf(x)));
}

// Fill the event-output tail: tev (inf), yev (inf), et (0), num_spikes (0).
__global__ void snn_init_tail(float* __restrict__ out) {
  float* tail = out + (size_t)TSTEPS * NN * 3;
  const size_t n_inf = (size_t)MAXSP + (size_t)MAXSP * NN * 3;   // tev + yev
  const size_t n_tot = n_inf + (size_t)MAXSP * NN + 1;           // + et + nsp
  size_t idx    = (size_t)blockIdx.x * blockDim.x + threadIdx.x;
  size_t stride = (size_t)gridDim.x * blockDim.x;
  for (size_t k = idx; k < n_tot; k += stride)
    tail[k] = (k < n_inf) ? __builtin_inff() : 0.0f;
}

// Persistent single-workgroup sequential scan: 32 wave32 waves on one WGP.
// Thread tid owns neurons j0=tid and j1=tid+1024; all state in registers.
__global__ void __launch_bounds__(1024)
snn_scan(const float* __restrict__ ts, const float* __restrict__ v0,
         const float* __restrict__ i0, const float* __restrict__ icp,
         const float* __restrict__ w,  const float* __restrict__ mu,
         const int* __restrict__ seedp, float* __restrict__ out) {
  __shared__ float    ts_s[TSTEPS];   // 16 KB
  __shared__ float    wrow[NN];       // 8 KB staging for w[eidx] broadcast
  __shared__ unsigned eidx_s;         // min spiking neuron index this step
  __shared__ unsigned nsp_s;          // spike-step counter

  const int tid = threadIdx.x;
  const int j0 = tid, j1 = tid + 1024;

  float* out_ys  = out;
  float* out_tev = out + (size_t)TSTEPS * NN * 3;
  float* out_yev = out_tev + MAXSP;
  float* out_et  = out_yev + (size_t)MAXSP * NN * 3;
  float* out_nsp = out_et + (size_t)MAXSP * NN;

  // Stage timestamps into LDS (low-latency per-step reads).
  for (int k = tid; k < TSTEPS; k += 1024) ts_s[k] = ts[k];

  // key = (0, seed); (k_init, k_trans) = jax.random.split(key)
  // split counts = iota(4): pairs (0,2),(1,3);
  // k_init = (f(0,2).a, f(1,3).a); k_trans = (f(0,2).b, f(1,3).b)
  const uint32_t seed = (uint32_t)seedp[0];
  uint32_t a0, a1, b0, b1;
  threefry2x32(0u, seed, 0u, 2u, a0, a1);
  threefry2x32(0u, seed, 1u, 3u, b0, b1);
  const uint32_t ki0 = a0, ki1 = b0;   // k_init
  const uint32_t kt0 = a1, kt1 = b1;   // k_trans

  // s0 = log(uniform(k_init, (N,))) - ALPHA.
  // uniform counts iota(2048) pair as (j, j+1024) -> one threefry per thread
  // yields bits for both owned neurons.
  uint32_t r0, r1;
  threefry2x32(ki0, ki1, (uint32_t)j0, (uint32_t)j1, r0, r1);
  float v[2]   = { v0[j0],  v0[j1]  };
  float cur[2] = { i0[j0],  i0[j1]  };
  float s[2]   = { logf(u01(r0)) - ALPHA_C, logf(u01(r1)) - ALPHA_C };
  const float icr[2] = { icp[j0], icp[j1] };
  const float mu0 = mu[0], mu1 = mu[1];

  // ys[0] = y0
  {
    size_t p0 = ((size_t)j0) * 3, p1 = ((size_t)j1) * 3;
    out_ys[p0 + 0] = v[0]; out_ys[p0 + 1] = cur[0]; out_ys[p0 + 2] = s[0];
    out_ys[p1 + 0] = v[1]; out_ys[p1 + 1] = cur[1]; out_ys[p1 + 2] = s[1];
  }
  if (tid == 0) { eidx_s = 0xFFFFFFFFu; nsp_s = 0u; }
  __syncthreads();

  for (int t = 0; t < TSTEPS - 1; ++t) {
    const float tnext = ts_s[t + 1];
    const float dt    = tnext - ts_s[t];

    // ---- Phase A: Euler drift + event detect + argmin reduction ----
    float v1[2], i1[2], s1[2];
    bool  ev[2];
#pragma unroll
    for (int q = 0; q < 2; ++q) {
      v1[q] = v[q]   + dt * (mu0 * (cur[q] + icr[q] - v[q]));
      i1[q] = cur[q] + dt * (-mu1 * cur[q]);
      s1[q] = s[q]   + dt * softplus(v[q]);
      ev[q] = (s1[q] >= 0.0f);
    }
    if (ev[0]) atomicMin(&eidx_s, (unsigned)j0);
    if (ev[1]) atomicMin(&eidx_s, (unsigned)j1);
    __syncthreads();                                   // B1

    const unsigned eidx = eidx_s;
    const bool     has  = (eidx != 0xFFFFFFFFu);
    const unsigned nsp  = nsp_s;

    // ---- Broadcast w[eidx] row into LDS via async DMA (ASYNCcnt path) ----
#if USE_ASYNC
    if (has && tid < 512) {
      const float* src = w + (size_t)eidx * NN + (size_t)tid * 4;
      __builtin_amdgcn_global_load_async_to_lds_b128(
          (g_v4i_ptr)(v4i_t*)(const_cast<float*>(src)),
          (l_v4i_ptr)(v4i_t*)(&wrow[tid * 4]),
          0, 0);
    }
    __builtin_amdgcn_s_wait_asynccnt(0);
#endif
    __syncthreads();                                   // B2

    // ---- Phase B: transition + state update ----
    if (has) {
      // k_step = fold_in(k_trans, t) = threefry(k_trans, (0, t));
      // u bits  = threefry(k_step, (j, j+1024))
      uint32_t kk0, kk1, q0, q1;
      threefry2x32(kt0, kt1, 0u, (uint32_t)t, kk0, kk1);
      threefry2x32(kk0, kk1, (uint32_t)j0, (uint32_t)j1, q0, q1);
      const float lu[2] = { logf(u01(q0)) - ALPHA_C, logf(u01(q1)) - ALPHA_C };
#pragma unroll
      for (int q = 0; q < 2; ++q) {
        const int j = q ? j1 : j0;
#if USE_ASYNC
        const float wv = wrow[j];
#else
        const float wv = w[(size_t)eidx * NN + j];
#endif
        v[q]   = v1[q] - (ev[q] ? VRESET : 0.0f);
        cur[q] = i1[q] + wv;
        s[q]   = ev[q] ? lu[q] : s1[q];
      }
    } else {
#pragma unroll
      for (int q = 0; q < 2; ++q) { v[q] = v1[q]; cur[q] = i1[q]; s[q] = s1[q]; }
    }

    // ys[t+1] = y2 (contiguous 24 KB store per step)
    {
      size_t p0 = ((size_t)(t + 1) * NN + j0) * 3;
      size_t p1 = ((size_t)(t + 1) * NN + j1) * 3;
      out_ys[p0 + 0] = v[0]; out_ys[p0 + 1] = cur[0]; out_ys[p0 + 2] = s[0];
      out_ys[p1 + 0] = v[1]; out_ys[p1 + 1] = cur[1]; out_ys[p1 + 2] = s[1];
    }

    // Event recording (first MAXSP spike steps record y1 / ev / t_next)
    if (has && nsp < MAXSP) {
      const unsigned widx = nsp;
      if (tid == 0) out_tev[widx] = tnext;
#pragma unroll
      for (int q = 0; q < 2; ++q) {
        const int j = q ? j1 : j0;
        const size_t eb = ((size_t)widx * NN + j) * 3;
        out_yev[eb + 0] = v1[q]; out_yev[eb + 1] = i1[q]; out_yev[eb + 2] = s1[q];
        out_et[(size_t)widx * NN + j] = ev[q] ? 1.0f : 0.0f;
      }
    }

    if (tid == 0) {
      if (has) nsp_s = nsp + 1u;
      eidx_s = 0xFFFFFFFFu;   // safe: all reads of eidx_s happened before B2
    }
    __syncthreads();                                   // B3
  }

  if (tid == 0) out_nsp[0] = (float)nsp_s;
}

extern "C" void kernel_launch(void* const* d_in, const int* in_sizes, int n_in,
                              void* d_out, int out_size, void* d_ws, size_t ws_size,
                              hipStream_t stream) {
  (void)in_sizes; (void)n_in; (void)out_size; (void)d_ws; (void)ws_size;
  const float* ts   = (const float*)d_in[0];
  const float* v0   = (const float*)d_in[1];
  const float* i0   = (const float*)d_in[2];
  const float* ic   = (const float*)d_in[3];
  const float* w    = (const float*)d_in[4];
  const float* mu   = (const float*)d_in[5];
  const int*   seed = (const int*)d_in[6];
  // d_in[7] = max_spikes (compile-time MAXSP)
  float* out = (float*)d_out;

  snn_init_tail<<<4096, 256, 0, stream>>>(out);
  snn_scan<<<1, 1024, 0, stream>>>(ts, v0, i0, ic, w, mu, seed, out);
}